// MXIntSoftmax_61314953118469
// MI455X (gfx1250) — compile-verified
//
#include <hip/hip_runtime.h>

#define D_COLS 2048
#define BLOCK 256
#define EPT (D_COLS / BLOCK)     // 8 floats = 32 bytes per thread
#define ROWS_PER_BLOCK 8
#define WAVES (BLOCK / 32)       // wave32 on gfx1250
#define NBUF 3                   // triple buffer: prefetch depth 2

typedef __attribute__((ext_vector_type(4))) float v4f;
typedef __attribute__((ext_vector_type(4))) int   v4i;
typedef __attribute__((address_space(1))) v4i* gv4i_t;   // global int4*
typedef __attribute__((address_space(3))) v4i* lv4i_t;   // LDS int4*
typedef __attribute__((address_space(1))) char* gchr_t;
typedef __attribute__((address_space(3))) char* lchr_t;

static __device__ __forceinline__ lv4i_t as_lds_v4(void* p) {
  // generic LDS addr: low 32 bits are the LDS byte offset (ISA 10.2 aperture rules)
  return (lv4i_t)(unsigned)(unsigned long long)p;
}
static __device__ __forceinline__ gv4i_t as_glob_v4(const void* p) {
  return (gv4i_t)(unsigned long long)p;
}

template <int Cnt>
static __device__ __forceinline__ void wait_asynccnt() {
#if __has_builtin(__builtin_amdgcn_s_wait_asynccnt)
  __builtin_amdgcn_s_wait_asynccnt(Cnt);
#else
  asm volatile("s_wait_asynccnt %0" ::"i"(Cnt) : "memory");
#endif
  asm volatile("" ::: "memory");  // keep LDS reads after the wait
}

// Async-copy 32 contiguous bytes per lane: global -> LDS (tracked by ASYNCcnt).
static __device__ __forceinline__ void async_copy_32B(void* lds_dst, const void* gsrc) {
#if __has_builtin(__builtin_amdgcn_global_load_async_to_lds_b128)
  __builtin_amdgcn_global_load_async_to_lds_b128(as_glob_v4(gsrc), as_lds_v4(lds_dst), 0, 0);
  __builtin_amdgcn_global_load_async_to_lds_b128(as_glob_v4(gsrc), as_lds_v4(lds_dst), 16, 0);
#else
  asm volatile("global_load_async_to_lds_b128 %0, %1, off\n\t"
               "global_load_async_to_lds_b128 %0, %1, off offset:16"
               ::"v"((lchr_t)(unsigned)(unsigned long long)lds_dst),
                 "v"((gchr_t)(unsigned long long)gsrc)
               : "memory");
#endif
}

// mxint_quant(x,8,4) -> y=qx*1.4375 -> floor-quant(9,5) -> (mexp in [64,127], eexp=n)
static __device__ __forceinline__ void quant_in(float xv, int& m_out, int& e_out) {
  float ax = __builtin_fabsf(xv);
  unsigned b = __float_as_uint(ax);
  int E = (int)(b >> 23) - 127;
  int e = ((b & 0x7fffffu) != 0u) ? E + 1 : E;   // ceil(log2(ax))
  e = (ax == 0.0f) ? 0 : e;
  e = e < -8 ? -8 : (e > 7 ? 7 : e);
  float m = rintf(ldexpf(xv, 7 - e));            // round(x * 2^-e * 128), RNE
  m = fminf(fmaxf(m, -128.0f), 127.0f);
  float qx = ldexpf(m, e - 7);
  float y = qx * 1.4375f;                        // quantized log2(e)
  float t = floorf(y * 32.0f);
  t = fminf(fmaxf(t, -256.0f), 255.0f);
  int ti = (int)t;
  int n = ti >> 5;                               // floor(y)
  int j = ti & 31;                               // r * 32
  float r = (float)j * 0.03125f;
  int mexp = (int)rintf(__builtin_amdgcn_exp2f(r) * 64.0f);  // v_exp_f32
  m_out = mexp > 127 ? 127 : mexp;
  e_out = n;
}

__global__ __launch_bounds__(BLOCK) void mxint_softmax_kernel(
    const float* __restrict__ x, float* __restrict__ out, int n_rows) {
  __shared__ __align__(16) float buf[NBUF][D_COLS];  // 24 KB triple buffer
  __shared__ int smax[WAVES];
  __shared__ int ssum[WAVES];

  const int tid  = (int)threadIdx.x;
  const int lane = tid & 31;
  const int wid  = tid >> 5;
  const int col  = tid * EPT;                      // this thread's 32B chunk
  const long long row0 = (long long)blockIdx.x * ROWS_PER_BLOCK;

  // prologue: prefetch rows 0 and 1 (depth-2 pipeline)
  if (row0 < n_rows)
    async_copy_32B(&buf[0][col], x + row0 * D_COLS + col);
  if (ROWS_PER_BLOCK > 1 && row0 + 1 < n_rows)
    async_copy_32B(&buf[1][col], x + (row0 + 1) * D_COLS + col);

  for (int r = 0; r < ROWS_PER_BLOCK; ++r) {
    const long long row = row0 + r;
    if (row >= n_rows) break;
    const int cur = r % NBUF;
    const bool p2 = (r + 2 < ROWS_PER_BLOCK) && (row + 2 < n_rows);
    const bool p1 = (r + 1 < ROWS_PER_BLOCK) && (row + 1 < n_rows);
    if (p2) {
      // slot (r+2)%3 held row r-1; this wave's reads of it finished last iter.
      async_copy_32B(&buf[(r + 2) % NBUF][col], x + (row + 2) * D_COLS + col);
      wait_asynccnt<4>();   // row r done; rows r+1, r+2 (2 instrs each) in flight
    } else if (p1) {
      wait_asynccnt<2>();   // row r done; row r+1 in flight
    } else {
      wait_asynccnt<0>();
    }

    // ---- pass 1: per-element quantize (each wave reads only its own chunk) ----
    v4f a = *(const v4f*)&buf[cur][col];
    v4f b = *(const v4f*)&buf[cur][col + 4];
    float xv[EPT] = {a.x, a.y, a.z, a.w, b.x, b.y, b.z, b.w};
    int me[EPT], ee[EPT];
    int lmax = -8;
#pragma unroll
    for (int i = 0; i < EPT; ++i) {
      quant_in(xv[i], me[i], ee[i]);
      lmax = ee[i] > lmax ? ee[i] : lmax;
    }

    // ---- block max of eexp (wave32 shuffle + LDS) ----
#pragma unroll
    for (int off = 16; off > 0; off >>= 1) {
      int o = __shfl_xor(lmax, off, 32);
      lmax = o > lmax ? o : lmax;
    }
    if (lane == 0) smax[wid] = lmax;
    __syncthreads();
    int emax = smax[0];
#pragma unroll
    for (int w = 1; w < WAVES; ++w) emax = smax[w] > emax ? smax[w] : emax;

    // ---- block sum of aligned mantissas: floor(mexp*16 * 2^(eexp-emax)) ----
    int lsum = 0;
#pragma unroll
    for (int i = 0; i < EPT; ++i) lsum += (me[i] << 4) >> (emax - ee[i]);
#pragma unroll
    for (int off = 16; off > 0; off >>= 1) lsum += __shfl_xor(lsum, off, 32);
    if (lane == 0) ssum[wid] = lsum;
    __syncthreads();
    int S = 0;
#pragma unroll
    for (int w = 0; w < WAVES; ++w) S += ssum[w];

    // ---- pass 2: mout = floor(mexp*256/S); rescale; re-quantize MXInt8 ----
    const float fS = (float)S;
    float o[EPT];
#pragma unroll
    for (int i = 0; i < EPT; ++i) {
      int num = me[i] << 8;                        // <= 32512, exact in fp32
      int q = (int)__fdividef((float)num, fS);     // fast divide
      if (q * S > num) --q;                        // floor fixup (q <= 31)
      else if ((q + 1) * S <= num) ++q;
      float res = 0.0f;
      if (q > 0) {
        float qo = ldexpf((float)q, ee[i] - emax - 4);
        unsigned bb = __float_as_uint(qo);
        int E2 = (int)(bb >> 23) - 127;
        int e2 = ((bb & 0x7fffffu) != 0u) ? E2 + 1 : E2;
        e2 = e2 < -8 ? -8 : (e2 > 7 ? 7 : e2);
        float m2 = rintf(ldexpf(qo, 7 - e2));
        m2 = fminf(m2, 127.0f);
        res = ldexpf(m2, e2 - 7);
      }
      o[i] = res;
    }
    v4f w0 = {o[0], o[1], o[2], o[3]};
    v4f w1 = {o[4], o[5], o[6], o[7]};
    float* op = out + row * D_COLS + col;
    __builtin_nontemporal_store(w0, (v4f*)op);     // write-once stream: th NT
    __builtin_nontemporal_store(w1, (v4f*)op + 1);
  }
}

extern "C" void kernel_launch(void* const* d_in, const int* in_sizes, int n_in,
                              void* d_out, int out_size, void* d_ws, size_t ws_size,
                              hipStream_t stream) {
  const float* x = (const float*)d_in[0];
  float* out = (float*)d_out;
  const int n_rows = in_sizes[0] / D_COLS;                          // 8192
  const int grid = (n_rows + ROWS_PER_BLOCK - 1) / ROWS_PER_BLOCK;  // 1024
  mxint_softmax_kernel<<<grid, BLOCK, 0, stream>>>(x, out, n_rows);
}